// SkipGram_10376640987530
// MI455X (gfx1250) — compile-verified
//
#include <hip/hip_runtime.h>
#include <hip/hip_bf16.h>

// Skip-gram negative-sampling loss, CDNA5 (gfx1250).
// Memory-bound gather problem (~1.2 GB @ 23.3 TB/s ~= 52 us floor).
// Each wave handles 16 batch elements; all 16 dot products per element run
// through V_WMMA_F32_16X16X4_F32, extracting only the diagonal of each 16x16
// product (FLOP waste irrelevant vs the matrix-pipe ceiling).
//
// K-permutation trick: dot products are invariant under K reordering, so A
// (global) and B (LDS-staged center*dropout) are both loaded as one b128 per
// lane feeding TWO wmma ops, instead of one b64 per wmma. Both operands use
// the same permuted K map, so the products are exact.

typedef __attribute__((ext_vector_type(2))) float v2f;
typedef __attribute__((ext_vector_type(8))) float v8f;

#define EMB     512
#define NEGS    15
#define BATCH   32768
#define GROUPS  (BATCH / 16)
#define DCHUNKS (EMB / 8)     // 64 double-chunks; each = 2 WMMA k-steps
#define DROP_P  0.1f
#define KEEP_SCALE (1.0f / (1.0f - DROP_P))

__device__ __forceinline__ float log_sigmoid(float x) {
    // logsigmoid(x) = min(x,0) - log(1 + exp(-|x|))  (numerically stable)
    return fminf(x, 0.0f) - __logf(1.0f + __expf(-fabsf(x)));
}

__global__ __launch_bounds__(32)
void sg_group_kernel(const int* __restrict__ center,
                     const int* __restrict__ context,
                     const int* __restrict__ noise_idx,
                     const float* __restrict__ drop_u,
                     const float* __restrict__ Wc,
                     const float* __restrict__ Wx,
                     float* __restrict__ partial) {
    // B-operand staging: [dchunk][lane] -> float4. 64*32*16B = 32 KB.
    __shared__ float4 lds_b[DCHUNKS * 32];

    const int g    = blockIdx.x;        // group of 16 batch elements
    const int lane = threadIdx.x;       // 0..31 (wave32)
    const int j    = lane & 15;         // batch-in-group / matrix row-col id
    const int kq   = (lane >> 4) * 4;   // lanes 16..31 take the upper quad
    const int b    = g * 16 + j;

    // ---- Stage B = center embedding * dropout-keep (permuted-K layout) ----
    {
        const int cidx = center[b];
        const float* crow = Wc + (size_t)cidx * EMB;
        const float* drow = drop_u + (size_t)b * EMB;
        #pragma unroll 4
        for (int d = 0; d < DCHUNKS; ++d) {
            const int off = 8 * d + kq;
            float4 w = *(const float4*)(crow + off);
            float4 u = *(const float4*)(drow + off);
            float4 o;
            o.x = w.x * ((u.x >= DROP_P) ? KEEP_SCALE : 0.0f);
            o.y = w.y * ((u.y >= DROP_P) ? KEEP_SCALE : 0.0f);
            o.z = w.z * ((u.z >= DROP_P) ? KEEP_SCALE : 0.0f);
            o.w = w.w * ((u.w >= DROP_P) ? KEEP_SCALE : 0.0f);
            lds_b[d * 32 + lane] = o;
        }
    }
    __syncthreads();

    // ---- 16 row types: r==0 -> context (positive), r>=1 -> negatives ----
    float lacc = 0.0f;  // per-lane loss; meaningful on lanes 0-7 and 24-31
    for (int r = 0; r < 16; ++r) {
        const int ridx = (r == 0) ? context[b] : noise_idx[b * NEGS + (r - 1)];
        const float* arow = Wx + (size_t)ridx * EMB;

        v8f acc = {};
        #pragma unroll 4
        for (int d = 0; d < DCHUNKS; ++d) {
            const int off = 8 * d + kq;
            float4 a4 = *(const float4*)(arow + off);   // b128: feeds 2 wmma
            float4 b4 = lds_b[d * 32 + lane];           // b128: feeds 2 wmma
            v2f av0; av0.x = a4.x; av0.y = a4.y;
            v2f bv0; bv0.x = b4.x; bv0.y = b4.y;
            acc = __builtin_amdgcn_wmma_f32_16x16x4_f32(
                false, av0, false, bv0, (short)0, acc, false, false);
            v2f av1; av1.x = a4.z; av1.y = a4.w;
            v2f bv1; bv1.x = b4.z; bv1.y = b4.w;
            acc = __builtin_amdgcn_wmma_f32_16x16x4_f32(
                false, av1, false, bv1, (short)0, acc, false, false);
        }

        // Diagonal extraction: D[v][v] at lane v, VGPR v (v=0..7);
        // D[8+v][8+v] at lane 24+v, VGPR v.
        float dot = 0.0f;
        #pragma unroll
        for (int v = 0; v < 8; ++v) {
            float x = acc[v];
            if (lane == v || lane == 24 + v) dot = x;
        }
        const float xin = (r == 0) ? dot : -dot;
        const float ls  = log_sigmoid(xin);
        if (lane < 8 || lane >= 24) lacc += ls;
    }

    // ---- Wave reduction over the 16 active lanes -> per-group partial ----
    #pragma unroll
    for (int s = 16; s >= 1; s >>= 1) lacc += __shfl_xor(lacc, s, 32);
    if (lane == 0) partial[g] = lacc;
}

__global__ __launch_bounds__(256)
void sg_reduce_kernel(const float* __restrict__ partial, float* __restrict__ out) {
    __shared__ float sdata[256];
    float s = 0.0f;
    for (int i = threadIdx.x; i < GROUPS; i += 256) s += partial[i];
    sdata[threadIdx.x] = s;
    __syncthreads();
    #pragma unroll
    for (int st = 128; st >= 1; st >>= 1) {
        if ((int)threadIdx.x < st) sdata[threadIdx.x] += sdata[threadIdx.x + st];
        __syncthreads();
    }
    if (threadIdx.x == 0) out[0] = -sdata[0] / (float)BATCH;
}

extern "C" void kernel_launch(void* const* d_in, const int* in_sizes, int n_in,
                              void* d_out, int out_size, void* d_ws, size_t ws_size,
                              hipStream_t stream) {
    const int*   center    = (const int*)d_in[0];
    const int*   context   = (const int*)d_in[1];
    const int*   noise_idx = (const int*)d_in[2];
    const float* drop_u    = (const float*)d_in[3];
    const float* Wc        = (const float*)d_in[4];
    const float* Wx        = (const float*)d_in[5];
    float*       partial   = (float*)d_ws;   // GROUPS floats = 8 KB
    float*       out       = (float*)d_out;

    sg_group_kernel<<<GROUPS, 32, 0, stream>>>(center, context, noise_idx,
                                               drop_u, Wc, Wx, partial);
    sg_reduce_kernel<<<1, 256, 0, stream>>>(partial, out);
}